// GroupedEmbedding_51247549776293
// MI455X (gfx1250) — compile-verified
//
#include <hip/hip_runtime.h>

// Grouped embedding gather for MI455X (gfx1250).
// Pure data movement: ~2 GiB total traffic -> HBM-bound (~92 us @ 23.3 TB/s).
// Strategy:
//  - one wave32 per embedding row: lane l moves bytes [16*l, 16*l+16) of the
//    512-byte row via global_load_b128 / global_store_b128 (full coalescing).
//  - 4 independent rows in flight per thread for memory-level parallelism.
//  - non-temporal stores for the 1 GiB write-once output and non-temporal
//    loads for the streaming index array, so the 192 MB L2 is preserved for
//    the 410 MB table working set (rows are reused ~2.6x on average).
//  - all per-lane offsets are unsigned 32-bit (max offset 12.8M / 268M < 2^31)
//    and the per-row table base is forced scalar via readfirstlane, so loads
//    and stores use the GVS mode (64-bit SGPR base + 32-bit VGPR offset)
//    instead of per-lane 64-bit address chains.
// WMMA is inapplicable (zero FLOPs); TDM gather mode would need per-4KB SGPR
// descriptor construction from dynamic indices plus an LDS round trip, which
// only adds overhead to an already bandwidth-saturating direct path.

typedef float f32x4 __attribute__((ext_vector_type(4)));

#define N_TABLES      8
#define NUM_EMB       100000
#define EMB_DIM       128
#define IDX_PER_TABLE 262144
#define TOTAL_ROWS    (N_TABLES * IDX_PER_TABLE)   // 2,097,152 rows
#define ROWS_PER_ITER 8                            // 256 threads / 32 lanes
#define UNROLL        4
#define ROWS_PER_BLOCK (ROWS_PER_ITER * UNROLL)    // 32 rows per block

__global__ __launch_bounds__(256)
void GroupedEmbedding_51247549776293_kernel(const int*   __restrict__ indices,
                                            const float* __restrict__ tables,
                                            float*       __restrict__ out)
{
    const unsigned lane = threadIdx.x & 31u;  // which float4 of the 512B row
    const unsigned wrow = threadIdx.x >> 5;   // wave within block
    const unsigned row0 = blockIdx.x * ROWS_PER_BLOCK + wrow;
    const unsigned lelt = lane * 4u;          // float offset of this lane's chunk

    // Phase 1: issue all index loads (streaming -> non-temporal).
    unsigned rows[UNROLL];
    unsigned inds[UNROLL];
#pragma unroll
    for (int i = 0; i < UNROLL; ++i) {
        rows[i] = row0 + (unsigned)(i * ROWS_PER_ITER);
        inds[i] = (unsigned)__builtin_nontemporal_load(indices + rows[i]);
    }

    // Phase 2: gathered table loads. Table id is wave-uniform -> scalar base
    // (SALU mul, GVS addressing); per-lane offset fits unsigned 32-bit.
    f32x4 v[UNROLL];
#pragma unroll
    for (int i = 0; i < UNROLL; ++i) {
        const int    table = __builtin_amdgcn_readfirstlane((int)(rows[i] >> 18));
        const float* tbase = tables + (size_t)table * (size_t)(NUM_EMB * EMB_DIM);
        const unsigned off = inds[i] * (unsigned)EMB_DIM + lelt;   // < 12.8M
        v[i] = *(const f32x4*)(tbase + off);
    }

    // Phase 3: streaming write-once output -> non-temporal b128 stores,
    // 32-bit offsets (max 268,435,452 < 2^31).
#pragma unroll
    for (int i = 0; i < UNROLL; ++i) {
        const unsigned dst = rows[i] * (unsigned)EMB_DIM + lelt;
        __builtin_nontemporal_store(v[i], (f32x4*)(out + dst));
    }
}

extern "C" void kernel_launch(void* const* d_in, const int* in_sizes, int n_in,
                              void* d_out, int out_size, void* d_ws, size_t ws_size,
                              hipStream_t stream)
{
    (void)in_sizes; (void)n_in; (void)out_size; (void)d_ws; (void)ws_size;

    const int*   indices = (const int*)  d_in[0];  // [8, 262144] indices
    const float* tables  = (const float*)d_in[1];  // [8, 100000, 128] fp32
    float*       out     = (float*)      d_out;    // [2097152, 128] fp32

    const int blocks = TOTAL_ROWS / ROWS_PER_BLOCK;  // 65,536 — exact cover
    GroupedEmbedding_51247549776293_kernel<<<blocks, 256, 0, stream>>>(indices, tables, out);
}